// EMM_GPU_48069273977568
// MI455X (gfx1250) — compile-verified
//
#include <hip/hip_runtime.h>
#include <cstddef>
#include <cstdint>

// ---------------------------------------------------------------------------
// Types for WMMA fragments (wave32, CDNA5)
// ---------------------------------------------------------------------------
typedef __attribute__((ext_vector_type(2))) float v2f;
typedef __attribute__((ext_vector_type(8))) float v8f;

#define ACT_NONE    0
#define ACT_SIGMOID 1
#define ACT_RELU    2
#define ACT_CLIP01  3

// ---------------------------------------------------------------------------
// Workspace layout (float offsets)
// ---------------------------------------------------------------------------
constexpr int WS_GF    = 0;      // sigmoid gate (focus)
constexpr int WS_GW    = 1;      // sigmoid gate (wide)
constexpr int WS_ADD   = 2;      // add gate
constexpr int WS_ERASE = 3;      // erase gate
constexpr int WS_FWF   = 8;      // 9  raw focused write filter
constexpr int WS_WWF   = 24;     // 49 raw wide write filter
constexpr int WS_FRF   = 80;     // 9  raw focused read filter
constexpr int WS_WRF   = 96;     // 49 raw wide read filter
constexpr int WS_MWF   = 160;    // 2860 = 130*22
constexpr int WS_MWW   = 3072;   // 3484 = 134*26
constexpr int WS_MEM   = 6592;   // 2560 = 128*20
constexpr int WS_FOC   = 9216;   // 2268 = 126*18
constexpr int WS_WIDE  = 11520;  // 1708 = 122*14
constexpr int WS_TMPF  = 13312;  // 2048 relu(lin(focused, W_cfr))
constexpr int WS_TMPW  = 15360;  // 2048 lin(wide, W_cwr)
constexpr int WS_PART1 = 17408;  // 8 * 6464 partials (launch 1)
constexpr int WS_PART2 = 69120;  // 4 * 4096 partials (launch 2)

// ---------------------------------------------------------------------------
// GEMV descriptor table
// ---------------------------------------------------------------------------
struct GemvDesc {
  const float* W;     // [N,K] row-major
  const float* bias;  // [N]
  const float* x;     // [K]
  float*       part;  // [kSplit][N] partial sums
  float*       y;     // [N] final output
  int N, K, act, kSplit, ksegR;  // ksegR: 4-aligned K-segment length
};
struct GemvArgs { GemvDesc d[10]; int nDesc; };

__device__ __forceinline__ float apply_act(float v, int act) {
  if (act == ACT_SIGMOID) return 1.0f / (1.0f + expf(-v));
  if (act == ACT_RELU)    return fmaxf(v, 0.0f);
  if (act == ACT_CLIP01)  return fminf(fmaxf(v, 0.0f), 1.0f);
  return v;
}

// ---------------------------------------------------------------------------
// WMMA-based GEMV partial kernel.
// One wave = one (descriptor, K-segment, 16-row block). All per-wave indexing
// is forced through readfirstlane so the k-loop is scalar-controlled and EXEC
// stays all-ones across every v_wmma (ISA requirement).
// A fragment: 16x4 fp32 tile of W (lanes 0-15: K0..K1; lanes 16-31: K2..K3).
// B fragment: column 0 = x[k..k+3] (lane 0 -> K0/K1, lane 16 -> K2/K3),
// all other columns zero -> D column 0 = W_block * x.
// ---------------------------------------------------------------------------
__global__ void wmma_gemv_partial(GemvArgs args) {
  const int lane = (int)(threadIdx.x & 31);
  // Wave-uniform by construction; readfirstlane makes it SGPR-provable.
  const int wave = __builtin_amdgcn_readfirstlane(
      (int)((blockIdx.x * blockDim.x + threadIdx.x) >> 5));

  // wave -> (desc, segment, row-block): pure scalar search
  int rem = wave, di = -1, nbN = 0;
  for (int i = 0; i < args.nDesc; ++i) {
    nbN = (args.d[i].N + 15) >> 4;
    const int tot = nbN * args.d[i].kSplit;
    if (rem < tot) { di = i; break; }
    rem -= tot;
  }
  if (di < 0) return;                      // scalar branch, EXEC untouched
  const GemvDesc g = args.d[di];
  const int seg  = rem / nbN;
  const int rb   = rem - seg * nbN;
  const int row0 = rb << 4;

  const int laneMod  = lane & 15;
  const int laneHalf = lane >> 4;

  const int kStart = seg * g.ksegR;        // scalar
  int len = g.ksegR;
  if (kStart + len > g.K) len = g.K - kStart;
  if (len < 0) len = 0;

  int rowc = row0 + laneMod;
  if (rowc > g.N - 1) rowc = g.N - 1;      // clamp to stay in-bounds
  const float* Wp = g.W + (size_t)rowc * (size_t)g.K + (size_t)kStart + (laneHalf << 1);
  const float* xp = g.x + (size_t)kStart + (laneHalf << 1);
  const bool col0 = (laneMod == 0);        // lanes 0 and 16 carry column n=0

  v8f acc = {};
#pragma unroll 4
  for (int k = 0; k < len; k += 4) {
    v2f a;  a.x = Wp[k];  a.y = Wp[k + 1];            // global_load_b64
    const float x0 = xp[k], x1 = xp[k + 1];           // broadcast-friendly b64
    v2f b;  b.x = col0 ? x0 : 0.0f;  b.y = col0 ? x1 : 0.0f;
    acc = __builtin_amdgcn_wmma_f32_16x16x4_f32(
        false, a, false, b, (short)0, acc, false, false);
  }

  // D column 0: lane 0 holds rows row0..row0+7 (VGPR 0-7),
  //             lane 16 holds rows row0+8..row0+15.
  if (col0) {
    const int mbase = row0 + (laneHalf << 3);
    float* pp = g.part + (size_t)seg * (size_t)g.N;
#pragma unroll
    for (int i = 0; i < 8; ++i) {
      const int r = mbase + i;
      if (r < g.N) pp[r] = acc[i];
    }
  }
}

// Sum K-segments, add bias, apply activation.
__global__ void gemv_reduce(GemvArgs args, int totalN) {
  int idx = (int)(blockIdx.x * blockDim.x + threadIdx.x);
  if (idx >= totalN) return;
  int rem = idx, di = -1;
  for (int i = 0; i < args.nDesc; ++i) {
    if (rem < args.d[i].N) { di = i; break; }
    rem -= args.d[i].N;
  }
  if (di < 0) return;
  const GemvDesc g = args.d[di];
  float s = g.bias[rem];
  for (int sg = 0; sg < g.kSplit; ++sg) s += g.part[(size_t)sg * g.N + rem];
  g.y[rem] = apply_act(s, g.act);
}

// ---------------------------------------------------------------------------
// Write phase: blend write filters, 3x3 + 7x7 valid convs, update memory.
// mem = memory + (conv3(mwf,fwf_b) + conv7(mww,wwf_b)) * (1-erase) + 2*add
// ---------------------------------------------------------------------------
__global__ void write_mem_kernel(const float* __restrict__ memory,
                                 const float* __restrict__ fwf0,
                                 const float* __restrict__ wwf0,
                                 float* __restrict__ ws) {
  __shared__ float fwf_b[9], wwf_b[49];
  const float gf = ws[WS_GF], gw = ws[WS_GW];
  const float addv = ws[WS_ADD], erase = ws[WS_ERASE];
  const int tid = (int)threadIdx.x;
  if (tid < 9)  fwf_b[tid] = ws[WS_FWF + tid] * gf + fwf0[tid] * (1.0f - gf);
  if (tid < 49) wwf_b[tid] = ws[WS_WWF + tid] * gw + wwf0[tid] * (1.0f - gw);
  __syncthreads();

  const float* mwf = ws + WS_MWF;   // [130][22]
  const float* mww = ws + WS_MWW;   // [134][26]
  float* mem = ws + WS_MEM;         // [128][20]
  const float ke = 1.0f - erase;

  for (int i = tid; i < 128 * 20; i += (int)blockDim.x) {
    const int r = i / 20, c = i % 20;
    float fw = 0.0f;
#pragma unroll
    for (int dy = 0; dy < 3; ++dy)
#pragma unroll
      for (int dx = 0; dx < 3; ++dx)
        fw += mwf[(r + dy) * 22 + (c + dx)] * fwf_b[dy * 3 + dx];
    float wv = 0.0f;
#pragma unroll
    for (int dy = 0; dy < 7; ++dy)
#pragma unroll
      for (int dx = 0; dx < 7; ++dx)
        wv += mww[(r + dy) * 26 + (c + dx)] * wwf_b[dy * 7 + dx];
    mem[i] = memory[i] + (fw + wv) * ke + 2.0f * addv;
  }
}

// ---------------------------------------------------------------------------
// Read phase: blend read filters, relu(conv3(mem)) and relu(conv7(mem)).
// ---------------------------------------------------------------------------
__global__ void read_conv_kernel(const float* __restrict__ frf0,
                                 const float* __restrict__ wrf0,
                                 float* __restrict__ ws) {
  __shared__ float frf_b[9], wrf_b[49];
  const float gf = ws[WS_GF], gw = ws[WS_GW];
  const int tid = (int)threadIdx.x;
  if (tid < 9)  frf_b[tid] = ws[WS_FRF + tid] * gf + frf0[tid] * (1.0f - gf);
  if (tid < 49) wrf_b[tid] = ws[WS_WRF + tid] * gw + wrf0[tid] * (1.0f - gw);
  __syncthreads();

  const float* mem = ws + WS_MEM;   // [128][20]
  float* focused = ws + WS_FOC;     // [126][18]
  float* wide    = ws + WS_WIDE;    // [122][14]

  for (int i = tid; i < 126 * 18; i += (int)blockDim.x) {
    const int r = i / 18, c = i % 18;
    float s = 0.0f;
#pragma unroll
    for (int dy = 0; dy < 3; ++dy)
#pragma unroll
      for (int dx = 0; dx < 3; ++dx)
        s += mem[(r + dy) * 20 + (c + dx)] * frf_b[dy * 3 + dx];
    focused[i] = fmaxf(s, 0.0f);
  }
  for (int i = tid; i < 122 * 14; i += (int)blockDim.x) {
    const int r = i / 14, c = i % 14;
    float s = 0.0f;
#pragma unroll
    for (int dy = 0; dy < 7; ++dy)
#pragma unroll
      for (int dx = 0; dx < 7; ++dx)
        s += mem[(r + dy) * 20 + (c + dx)] * wrf_b[dy * 7 + dx];
    wide[i] = fmaxf(s, 0.0f);
  }
}

// read = lin(wide,W_cwr,b_cwr) + relu(lin(focused,W_cfr,b_cfr))
__global__ void combine_kernel(const float* __restrict__ ws,
                               float* __restrict__ out) {
  const int i = (int)(blockIdx.x * blockDim.x + threadIdx.x);
  if (i < 2048) out[i] = ws[WS_TMPW + i] + ws[WS_TMPF + i];
}

// ---------------------------------------------------------------------------
// Host side
// ---------------------------------------------------------------------------
static inline int ksegAligned(int K, int kSplit) {
  int ks = (K + kSplit - 1) / kSplit;
  return (ks + 3) & ~3;
}

extern "C" void kernel_launch(void* const* d_in, const int* in_sizes, int n_in,
                              void* d_out, int out_size, void* d_ws, size_t ws_size,
                              hipStream_t stream) {
  (void)in_sizes; (void)n_in; (void)out_size; (void)ws_size;
  const float* h_t    = (const float*)d_in[0];
  const float* memory = (const float*)d_in[1];
  const float* frf0   = (const float*)d_in[2];
  const float* wrf0   = (const float*)d_in[3];
  const float* fwf0   = (const float*)d_in[4];
  const float* wwf0   = (const float*)d_in[5];
  const float* W_fg = (const float*)d_in[6],  *b_fg = (const float*)d_in[7];
  const float* W_wg = (const float*)d_in[8],  *b_wg = (const float*)d_in[9];
  const float* W_rf = (const float*)d_in[10], *b_rf = (const float*)d_in[11];
  const float* W_rw = (const float*)d_in[12], *b_rw = (const float*)d_in[13];
  const float* W_cfr = (const float*)d_in[14], *b_cfr = (const float*)d_in[15];
  const float* W_cwr = (const float*)d_in[16], *b_cwr = (const float*)d_in[17];
  const float* W_hfc = (const float*)d_in[18], *b_hfc = (const float*)d_in[19];
  const float* W_hwc = (const float*)d_in[20], *b_hwc = (const float*)d_in[21];
  const float* W_ag = (const float*)d_in[22], *b_ag = (const float*)d_in[23];
  const float* W_eg = (const float*)d_in[24], *b_eg = (const float*)d_in[25];
  const float* W_wf = (const float*)d_in[26], *b_wf = (const float*)d_in[27];
  const float* W_ww = (const float*)d_in[28], *b_ww = (const float*)d_in[29];

  float* ws = (float*)d_ws;
  float* out = (float*)d_out;

  // ---- Launch 1: all h_t GEMVs (K = 4096), kSplit = 8 ----
  const int KS1 = 8;
  GemvArgs a1{};
  a1.nDesc = 10;
  const float* Ws1[10]  = {W_fg, W_wg, W_ag, W_eg, W_wf, W_ww, W_rf, W_rw, W_hfc, W_hwc};
  const float* Bs1[10]  = {b_fg, b_wg, b_ag, b_eg, b_wf, b_ww, b_rf, b_rw, b_hfc, b_hwc};
  const int    Ns1[10]  = {1, 1, 1, 1, 9, 49, 9, 49, 2860, 3484};
  const int    Ac1[10]  = {ACT_SIGMOID, ACT_SIGMOID, ACT_CLIP01, ACT_CLIP01,
                           ACT_CLIP01, ACT_CLIP01, ACT_CLIP01, ACT_CLIP01,
                           ACT_RELU, ACT_RELU};
  float* Ys1[10] = {ws + WS_GF, ws + WS_GW, ws + WS_ADD, ws + WS_ERASE,
                    ws + WS_FWF, ws + WS_WWF, ws + WS_FRF, ws + WS_WRF,
                    ws + WS_MWF, ws + WS_MWW};
  {
    int cumN = 0, waves = 0, totalN = 0;
    for (int i = 0; i < 10; ++i) {
      a1.d[i].W = Ws1[i]; a1.d[i].bias = Bs1[i]; a1.d[i].x = h_t;
      a1.d[i].y = Ys1[i]; a1.d[i].N = Ns1[i]; a1.d[i].K = 4096;
      a1.d[i].act = Ac1[i]; a1.d[i].kSplit = KS1;
      a1.d[i].ksegR = ksegAligned(4096, KS1);               // 512
      a1.d[i].part = ws + WS_PART1 + (size_t)KS1 * cumN;
      cumN += Ns1[i]; totalN += Ns1[i];
      waves += ((Ns1[i] + 15) >> 4) * KS1;
    }
    const int wavesPerBlock = 4;                            // 128 threads
    const int grid = (waves + wavesPerBlock - 1) / wavesPerBlock;
    wmma_gemv_partial<<<grid, 128, 0, stream>>>(a1);
    gemv_reduce<<<(totalN + 255) / 256, 256, 0, stream>>>(a1, totalN);
  }

  // ---- Write memory (tiny convs) ----
  write_mem_kernel<<<1, 256, 0, stream>>>(memory, fwf0, wwf0, ws);

  // ---- Read convs ----
  read_conv_kernel<<<1, 256, 0, stream>>>(frf0, wrf0, ws);

  // ---- Launch 2: read GEMVs, kSplit = 4 ----
  const int KS2 = 4;
  GemvArgs a2{};
  a2.nDesc = 2;
  a2.d[0] = {W_cfr, b_cfr, ws + WS_FOC, ws + WS_PART2, ws + WS_TMPF,
             2048, 2268, ACT_RELU, KS2, ksegAligned(2268, KS2)};
  a2.d[1] = {W_cwr, b_cwr, ws + WS_WIDE, ws + WS_PART2 + (size_t)KS2 * 2048,
             ws + WS_TMPW, 2048, 1708, ACT_NONE, KS2, ksegAligned(1708, KS2)};
  {
    const int waves = (((2048 + 15) >> 4) * KS2) * 2;       // 1024
    const int grid = (waves + 3) / 4;
    wmma_gemv_partial<<<grid, 128, 0, stream>>>(a2);
    gemv_reduce<<<(4096 + 255) / 256, 256, 0, stream>>>(a2, 4096);
  }

  // ---- Combine ----
  combine_kernel<<<(2048 + 255) / 256, 256, 0, stream>>>(ws, out);
}